// Glm4MoeNaiveMoe_1657857376737
// MI455X (gfx1250) — compile-verified
//
#include <hip/hip_runtime.h>
#include <hip/hip_bf16.h>

// ---------------------------------------------------------------------------
// Glm4MoeNaiveMoe for MI455X (gfx1250): routed grouped-GEMM MoE, fp32 WMMA,
// TDM gather-mode staging of the token tile into LDS.
//   T=1024 tokens, H=1024 hidden, I=1024 intermediate, E=16 experts, K=8 topk
// ---------------------------------------------------------------------------

#define T_TOK 1024
#define H_DIM 1024
#define I_DIM 1024
#define E_EXP 16
#define TOPK  8

#define TM        16                   // tokens per tile (WMMA M)
#define CHUNK     256                  // I-chunk fused between GEMM1 and GEMM2
#define HS_PAD    (H_DIM + 4)          // 1028: TDM tile_dim0 over-read -> padded rows
#define ACT_PAD   (CHUNK + 4)
#define PAD_TOTAL (T_TOK * TOPK + E_EXP * TM)   // 8448 aligned slots max
#define NTILES    (PAD_TOTAL / TM)              // 528

typedef float v2f __attribute__((ext_vector_type(2)));
typedef float v8f __attribute__((ext_vector_type(8)));
typedef unsigned int tdm_u32x4 __attribute__((ext_vector_type(4)));
typedef int          tdm_i32x8 __attribute__((ext_vector_type(8)));
typedef int          tdm_i32x4 __attribute__((ext_vector_type(4)));

// ---------------------------------------------------------------------------
__global__ void zero_out_kernel(float* __restrict__ out, int n) {
    int i = blockIdx.x * blockDim.x + threadIdx.x;
    if (i < n) out[i] = 0.0f;
}

// ---------------------------------------------------------------------------
// Routing: build per-expert, 16-aligned slot lists of (token, weight).
// Padding slots get weight 0 -> contribute nothing. Deterministic assignment;
// per-expert token lists come out non-decreasing (TDM gather OOB requirement).
__global__ void moe_route_kernel(const int* __restrict__ topk_idx,   // [T,K]
                                 const float* __restrict__ topk_w,   // [T,K]
                                 int* __restrict__ offA,             // [E+1]
                                 int* __restrict__ tok,              // [PAD_TOTAL]
                                 float* __restrict__ wgt) {          // [PAD_TOTAL]
    __shared__ int s_cnt[E_EXP];
    __shared__ int s_off[E_EXP + 1];
    const int tid = threadIdx.x;

    if (tid < E_EXP) s_cnt[tid] = 0;
    __syncthreads();

    for (int i = tid; i < T_TOK * TOPK; i += blockDim.x) {
        int e = topk_idx[i];
        atomicAdd(&s_cnt[e], 1);
    }
    __syncthreads();

    if (tid == 0) {
        int o = 0;
        for (int e = 0; e < E_EXP; ++e) {
            s_off[e] = o;
            o += (s_cnt[e] + TM - 1) & ~(TM - 1);   // align each expert to 16
        }
        s_off[E_EXP] = o;
    }
    __syncthreads();

    // zero all slots (covers padding), then fill real entries
    for (int i = tid; i < PAD_TOTAL; i += blockDim.x) { tok[i] = 0; wgt[i] = 0.0f; }
    if (tid <= E_EXP) offA[tid] = s_off[tid];
    __syncthreads();

    if (tid < E_EXP) {          // thread e owns expert e -> deterministic order
        int slot = s_off[tid];
        for (int i = 0; i < T_TOK * TOPK; ++i) {
            if (topk_idx[i] == tid) {
                tok[slot] = i / TOPK;
                wgt[slot] = topk_w[i];
                ++slot;
            }
        }
    }
}

// ---------------------------------------------------------------------------
// Fused expert tile kernel: one block = 16 routed token-slots of one expert.
//   Stage:  TDM gather (tensor_load_to_lds) of the 16 token rows -> LDS
//   GEMM1:  gate/up = hs[16,H] x Wgu[e]^T   (v_wmma_f32_16x16x4_f32)
//   act   = silu(gate)*up * token_weight    (weight folded in -> linear)
//   GEMM2:  out[16,H] += act[16,I] x Wd[e]^T, atomically combined into d_out
__launch_bounds__(256)
__global__ void moe_expert_kernel(const float* __restrict__ hs,     // [T,H]
                                  const float* __restrict__ gup,    // [E,2I,H]
                                  const float* __restrict__ dwn,    // [E,H,I]
                                  const int* __restrict__ offA,     // [E+1]
                                  const int* __restrict__ tok,      // slots
                                  const float* __restrict__ wgt,    // slots
                                  float* __restrict__ out) {        // [T,H]
    __shared__ float s_hs[TM * HS_PAD];     // 16 x 1028 f32  (~64.3 KB)
    __shared__ float s_act[TM * ACT_PAD];   // 16 x 260  f32  (~16.3 KB)
    __shared__ int   s_tok[TM];
    __shared__ float s_w[TM];
    __shared__ int   s_exp;

    const int tid      = threadIdx.x;
    const int tileBase = blockIdx.x * TM;

    if (tid == 0) {
        int e = -1;
        if (tileBase < offA[E_EXP]) {
            for (int i = 0; i < E_EXP; ++i)
                if (tileBase >= offA[i] && tileBase < offA[i + 1]) { e = i; break; }
        }
        s_exp = e;
    }
    __syncthreads();
    const int e = s_exp;
    if (e < 0) return;                       // slack tiles past total routed work

    if (tid < TM) { s_tok[tid] = tok[tileBase + tid]; s_w[tid] = wgt[tileBase + tid]; }
    __syncthreads();

    const int lane = tid & 31;
    const int wave = tid >> 5;
    const int half = lane >> 4;              // lane half selects K/M offset
    const int mn   = lane & 15;              // A: row M ; B/C/D: col N

#if __has_builtin(__builtin_amdgcn_tensor_load_to_lds) && \
    __has_builtin(__builtin_amdgcn_s_wait_tensorcnt)
    // ---- TDM gather: 16 scattered token rows -> LDS, 1028-dword padded rows.
    // tile_dim0 = 1028 > tensor_dim0 = 1024: the 4 OOB dwords per row read as
    // zero, giving conflict-free column access (row stride 1028 % 64 == 4).
    if (wave == 0) {
        const uint32_t laddr = (uint32_t)(uintptr_t)&s_hs[0];
        const uint64_t gaddr = (uint64_t)(uintptr_t)hs;

        int p[8];
        #pragma unroll
        for (int j = 0; j < 8; ++j)
            p[j] = __builtin_amdgcn_readfirstlane(
                (s_tok[2 * j] & 0xFFFF) | (s_tok[2 * j + 1] << 16));

        // Group0: count=1 | gather_mode(16-bit idx) ; lds_addr ; global_addr ; type=2
        tdm_u32x4 g0 = { 0x80000001u,
                         laddr,
                         (uint32_t)gaddr,
                         ((uint32_t)(gaddr >> 32) & 0x01FFFFFFu) | 0x80000000u };
        // Group1: data_size=4B | dim0=1024 | dim1=1024 | tile0=1028 | tile1=16 idx
        //         | dim0_stride=1024
        tdm_i32x8 g1 = { 0x00020000,      // [17:16] data_size = 2 (4 bytes)
                         0x04000000,      // [63:48] tensor_dim0.lo16 = 1024
                         0x04000000,      // [95:80] tensor_dim1.lo16 = 1024
                         0x04040000,      // [127:112] tile_dim0 = 1028
                         16,              // [143:128] tile_dim1 = 16 row indices
                         1024,            // [191:160] tensor_dim0_stride = 1024
                         0, 0 };
        tdm_i32x4 g2 = { p[0], p[1], p[2], p[3] };   // row indices 0..7
        tdm_i32x4 g3 = { p[4], p[5], p[6], p[7] };   // row indices 8..15
        tdm_i32x8 gx = { 0, 0, 0, 0, 0, 0, 0, 0 };   // unused trailing group

        __builtin_amdgcn_tensor_load_to_lds(g0, g1, g2, g3, gx, 0);
        __builtin_amdgcn_s_wait_tensorcnt(0);        // TENSORcnt == 0
    }
    __syncthreads();
#else
    // Fallback: cooperative float4 staging (row-end padding only).
    for (int r = 0; r < TM; ++r) {
        const float4 v = ((const float4*)(hs + (size_t)s_tok[r] * H_DIM))[tid];
        *(float4*)&s_hs[r * HS_PAD + tid * 4] = v;
    }
    __syncthreads();
#endif

    const float* Wg = gup + (size_t)e * (2 * I_DIM * H_DIM);
    const float* Wd = dwn + (size_t)e * (H_DIM * I_DIM);

    v8f acc[8];                              // 8 H-tiles per wave, K = full I
    #pragma unroll
    for (int jt = 0; jt < 8; ++jt)
        #pragma unroll
        for (int r = 0; r < 8; ++r) acc[jt][r] = 0.0f;

    for (int c = 0; c < I_DIM; c += CHUNK) {
        __syncthreads();                     // previous GEMM2 finished with s_act

        // ---- GEMM1 + SiLU for this I-chunk: wave handles i-tiles {wave, wave+8}
        for (int lt = wave; lt < CHUNK / 16; lt += 8) {
            const int    i0    = c + lt * 16;
            const float* growg = Wg + (size_t)(i0 + mn) * H_DIM;          // gate row
            const float* growu = Wg + (size_t)(I_DIM + i0 + mn) * H_DIM;  // up row

            if (lt < 8) {                    // prefetch next i-tile's weight rows
                __builtin_prefetch(growg + (size_t)128 * H_DIM, 0, 1);
                __builtin_prefetch(growu + (size_t)128 * H_DIM, 0, 1);
            }

            v8f dg = {0.f,0.f,0.f,0.f,0.f,0.f,0.f,0.f};
            v8f du = {0.f,0.f,0.f,0.f,0.f,0.f,0.f,0.f};

            for (int k = 0; k < H_DIM; k += 4) {
                const int ko = k + 2 * half;                 // K pair per lane half
                v2f a  = *(const v2f*)&s_hs[mn * HS_PAD + ko];
                v2f bg = *(const v2f*)(growg + ko);
                v2f bu = *(const v2f*)(growu + ko);
                dg = __builtin_amdgcn_wmma_f32_16x16x4_f32(false, a, false, bg,
                                                           (short)0, dg, false, false);
                du = __builtin_amdgcn_wmma_f32_16x16x4_f32(false, a, false, bu,
                                                           (short)0, du, false, false);
            }
            #pragma unroll
            for (int r = 0; r < 8; ++r) {
                const int   m = r + 8 * half;                // C/D row for this reg
                const float g = dg[r];
                const float u = du[r];
                const float s = g / (1.0f + __expf(-g));     // silu(gate)
                s_act[m * ACT_PAD + lt * 16 + mn] = s * u * s_w[m];
            }
        }
        __syncthreads();

        // ---- GEMM2 partial: accumulate this chunk's K into 8 H-tiles per wave
        for (int kk = 0; kk < CHUNK; kk += 4) {
            const int ko = kk + 2 * half;
            v2f a = *(const v2f*)&s_act[mn * ACT_PAD + ko];
            #pragma unroll
            for (int jt = 0; jt < 8; ++jt) {
                const int    h0   = wave * 128 + jt * 16;
                const float* drow = Wd + (size_t)(h0 + mn) * I_DIM + c;
                v2f b = *(const v2f*)(drow + ko);
                acc[jt] = __builtin_amdgcn_wmma_f32_16x16x4_f32(false, a, false, b,
                                                                (short)0, acc[jt],
                                                                false, false);
            }
        }
    }

    // ---- Weighted combine into d_out (tokens collide across experts -> atomic)
    #pragma unroll
    for (int jt = 0; jt < 8; ++jt) {
        const int h0 = wave * 128 + jt * 16;
        #pragma unroll
        for (int r = 0; r < 8; ++r) {
            const int m = r + 8 * half;
            const int t = s_tok[m];
            atomicAdd(&out[(size_t)t * H_DIM + h0 + mn], acc[jt][r]);
        }
    }
}

// ---------------------------------------------------------------------------
extern "C" void kernel_launch(void* const* d_in, const int* in_sizes, int n_in,
                              void* d_out, int out_size, void* d_ws, size_t ws_size,
                              hipStream_t stream) {
    const float* hs   = (const float*)d_in[0];   // hidden_states [T,H]
    const int*   idx  = (const int*)  d_in[1];   // top_k_index   [T,K]
    const float* tkw  = (const float*)d_in[2];   // top_k_weights [T,K]
    const float* gup  = (const float*)d_in[3];   // gate_up_proj  [E,2I,H]
    const float* dwn  = (const float*)d_in[4];   // down_proj     [E,H,I]
    float*       out  = (float*)d_out;           // [T,H]

    // workspace layout: offA[32] | tok[PAD_TOTAL] | wgt[PAD_TOTAL]  (~66 KB)
    int*   offA = (int*)d_ws;
    int*   tok  = offA + 32;
    float* wgt  = (float*)(tok + PAD_TOTAL);

    zero_out_kernel<<<(T_TOK * H_DIM + 255) / 256, 256, 0, stream>>>(out, T_TOK * H_DIM);
    moe_route_kernel<<<1, 256, 0, stream>>>(idx, tkw, offA, tok, wgt);
    moe_expert_kernel<<<NTILES, 256, 0, stream>>>(hs, gup, dwn, offA, tok, wgt, out);
}